// PersonalizedCalibrationNetwork_78039555768660
// MI455X (gfx1250) — compile-verified
//
#include <hip/hip_runtime.h>

#define B_TOT 2048
#define IN_F  256
#define KDIM  257     // IN+1 (also L1+1, L2+1)
#define KPAD  260     // pad K to multiple of 4
#define QN    16
#define AN    4
#define JN    64

typedef float v2f __attribute__((ext_vector_type(2)));
typedef float v8f __attribute__((ext_vector_type(8)));

__device__ __forceinline__ float fsigmoid(float v) {
    // v_exp_f32 + v_rcp_f32 (1-ULP), avoids the IEEE div_scale/fma expansion
    return __builtin_amdgcn_rcpf(1.0f + __expf(-v));
}

// One dense layer for a 16-sample tile:
//   out_t[16][0..255] = sigmoid( in_t[16][0..256] @ (Wb + Wa)[257][256] )
// Each wave owns two 16-column N-tiles. K is processed in chunks of 16 rows
// (4 WMMA steps): all 32 global weight loads of a chunk are issued first into
// registers (one clause, partial-count waits), then the W+Wa fuses and 8 WMMAs
// run between waits. Bias row (k=256) is a peeled, select-based step.
__device__ __forceinline__ void dense_layer(
    const float* __restrict__ Wb, const float* __restrict__ Wa,
    const float (*in_t)[KPAD], float (*out_t)[KPAD],
    int wave, int hi, int lm)
{
    const int n0 = wave * 32 + lm;
    const int n1 = n0 + 16;
    const float* __restrict__ wb0 = Wb + n0;   // column bases
    const float* __restrict__ wa0 = Wa + n0;
    const float* __restrict__ wb1 = Wb + n1;
    const float* __restrict__ wa1 = Wa + n1;
    const int krow = 2 * hi;                   // this lane-half's K offset
    v8f acc0 = {};
    v8f acc1 = {};

    for (int kb = 0; kb < 256; kb += 16) {     // 4 WMMA steps per chunk
        float w[32];                           // raw loads, adds deferred
        #pragma unroll
        for (int s = 0; s < 4; ++s) {
            const int k0 = kb + 4 * s + krow;
            const int k1 = k0 + 1;
            w[s * 8 + 0] = wb0[k0 * 256];
            w[s * 8 + 1] = wa0[k0 * 256];
            w[s * 8 + 2] = wb0[k1 * 256];
            w[s * 8 + 3] = wa0[k1 * 256];
            w[s * 8 + 4] = wb1[k0 * 256];
            w[s * 8 + 5] = wa1[k0 * 256];
            w[s * 8 + 6] = wb1[k1 * 256];
            w[s * 8 + 7] = wa1[k1 * 256];
        }
        v2f av[4];
        #pragma unroll
        for (int s = 0; s < 4; ++s) {
            const int k0 = kb + 4 * s + krow;
            av[s].x = in_t[lm][k0];
            av[s].y = in_t[lm][k0 + 1];
        }
        #pragma unroll
        for (int s = 0; s < 4; ++s) {
            v2f b0, b1;
            b0.x = w[s * 8 + 0] + w[s * 8 + 1];
            b0.y = w[s * 8 + 2] + w[s * 8 + 3];
            b1.x = w[s * 8 + 4] + w[s * 8 + 5];
            b1.y = w[s * 8 + 6] + w[s * 8 + 7];
            acc0 = __builtin_amdgcn_wmma_f32_16x16x4_f32(false, av[s], false, b0,
                                                         (short)0, acc0, false, false);
            acc1 = __builtin_amdgcn_wmma_f32_16x16x4_f32(false, av[s], false, b1,
                                                         (short)0, acc1, false, false);
        }
    }
    {   // peeled tail: kb = 256 (bias row); rows 257..259 contribute zero
        v2f a;
        a.x = in_t[lm][256 + krow];            // hi=0: bias col (=1); hi=1: pad (=0)
        a.y = in_t[lm][257 + krow];            // pad cols are zero
        const float t0 = wb0[256 * 256] + wa0[256 * 256];  // valid for all lanes
        const float t1 = wb1[256 * 256] + wa1[256 * 256];
        v2f b0, b1;
        b0.x = hi ? 0.0f : t0;  b0.y = 0.0f;   // v_cndmask, no branch
        b1.x = hi ? 0.0f : t1;  b1.y = 0.0f;
        acc0 = __builtin_amdgcn_wmma_f32_16x16x4_f32(false, a, false, b0,
                                                     (short)0, acc0, false, false);
        acc1 = __builtin_amdgcn_wmma_f32_16x16x4_f32(false, a, false, b1,
                                                     (short)0, acc1, false, false);
    }
    #pragma unroll
    for (int r = 0; r < 8; ++r) {
        out_t[r + 8 * hi][n0] = fsigmoid(acc0[r]);
        out_t[r + 8 * hi][n1] = fsigmoid(acc1[r]);
    }
}

__global__ __launch_bounds__(256)
void pcn_kernel(const float* __restrict__ X,
                const int*   __restrict__ judges,
                const float* __restrict__ W1,
                const float* __restrict__ W1a,
                const float* __restrict__ W2,
                const float* __restrict__ W2a,
                const float* __restrict__ V,
                const float* __restrict__ Va,
                float*       __restrict__ out)
{
    __shared__ int   s_idx[B_TOT];
    __shared__ int   s_ng;
    __shared__ float s_x[16][KPAD];   // x tile / z2 tile (bias cols persist)
    __shared__ float s_z[16][KPAD];   // z1 tile

    const int g    = blockIdx.x;      // judge
    const int t0   = blockIdx.y;      // tile slice
    const int tstr = gridDim.y;
    const int tid  = threadIdx.x;
    const int lane = tid & 31;
    const int wave = tid >> 5;
    const int hi   = lane >> 4;       // lane half (K offset selector)
    const int lm   = lane & 15;       // row / column within fragment

    // ---- group samples by judge: wave-0 ballot compaction (deterministic) ----
    if (wave == 0) {
        int count = 0;
        for (int base = 0; base < B_TOT; base += 32) {
            const bool m = (judges[base + lane] == g);
            const unsigned mask = (unsigned)__ballot(m);
            const int rank = __popc(mask & ((1u << lane) - 1u));
            if (m) s_idx[count + rank] = base + lane;
            count += __popc(mask);
        }
        if (lane == 0) s_ng = count;
    }
    __syncthreads();

    const int ng = s_ng;
    const float* W1g = W1a + (size_t)g * KDIM * 256;
    const float* W2g = W2a + (size_t)g * KDIM * 256;
    const float* Vg  = Va  + (size_t)g * QN * KDIM * AN;

    for (int tile = t0; tile * 16 < ng; tile += tstr) {
        // ---- gather 16 sample rows into LDS; col 256 = 1 (bias), 257..259 = 0
        {
            const int row  = tid >> 4;     // 0..15
            const int c0   = tid & 15;
            const int srow = tile * 16 + row;
            const bool act = srow < ng;
            const float* xp = act ? (X + (size_t)s_idx[srow] * IN_F) : X;
            for (int c = c0; c < IN_F; c += 16)
                s_x[row][c] = act ? xp[c] : 0.0f;
            if (c0 < 4)
                s_x[row][IN_F + c0] = (c0 == 0) ? 1.0f : 0.0f;
        }
        __syncthreads();

        // ---- layer 1: s_z = sigmoid(s_x @ (W1 + W1a[g])) ----
        dense_layer(W1, W1g, s_x, s_z, wave, hi, lm);
        if (tid < 64) {                    // bias/pad columns for z1b
            const int row = tid >> 2, c = tid & 3;
            s_z[row][IN_F + c] = (c == 0) ? 1.0f : 0.0f;
        }
        __syncthreads();

        // ---- layer 2: s_x = sigmoid(s_z @ (W2 + W2a[g]))  (bias cols persist)
        dense_layer(W2, W2g, s_z, s_x, wave, hi, lm);
        __syncthreads();

        // ---- layer 3: logits[b][q][a] = z2b @ (V + Va[g]); 64 cols = 4 N-tiles
        if (wave < 4) {
            const int n  = wave * 16 + lm;      // flattened (q,a) column, 0..63
            const int q  = n >> 2;
            const int ai = n & 3;
            const float* __restrict__ Vb  = V  + (size_t)q * KDIM * AN + ai;
            const float* __restrict__ Vaj = Vg + (size_t)q * KDIM * AN + ai;
            const int krow = 2 * hi;
            v8f acc = {};
            for (int kb = 0; kb < 256; kb += 16) {   // 4 WMMA steps per chunk
                float w[16];
                #pragma unroll
                for (int s = 0; s < 4; ++s) {
                    const int k0 = kb + 4 * s + krow;
                    const int k1 = k0 + 1;
                    w[s * 4 + 0] = Vb[k0 * AN];
                    w[s * 4 + 1] = Vaj[k0 * AN];
                    w[s * 4 + 2] = Vb[k1 * AN];
                    w[s * 4 + 3] = Vaj[k1 * AN];
                }
                v2f av[4];
                #pragma unroll
                for (int s = 0; s < 4; ++s) {
                    const int k0 = kb + 4 * s + krow;
                    av[s].x = s_x[lm][k0];
                    av[s].y = s_x[lm][k0 + 1];
                }
                #pragma unroll
                for (int s = 0; s < 4; ++s) {
                    v2f b;
                    b.x = w[s * 4 + 0] + w[s * 4 + 1];
                    b.y = w[s * 4 + 2] + w[s * 4 + 3];
                    acc = __builtin_amdgcn_wmma_f32_16x16x4_f32(false, av[s], false, b,
                                                                (short)0, acc, false, false);
                }
            }
            {   // peeled bias row k = 256
                v2f a;
                a.x = s_x[lm][256 + krow];
                a.y = s_x[lm][257 + krow];
                const float t = Vb[256 * AN] + Vaj[256 * AN];
                v2f b;
                b.x = hi ? 0.0f : t;
                b.y = 0.0f;
                acc = __builtin_amdgcn_wmma_f32_16x16x4_f32(false, a, false, b,
                                                            (short)0, acc, false, false);
            }
            #pragma unroll
            for (int r = 0; r < 8; ++r) {
                const int row  = r + 8 * hi;
                const int srow = tile * 16 + row;
                if (srow < ng)
                    out[(size_t)s_idx[srow] * (QN * AN) + n] = acc[r];
            }
        }
        __syncthreads();   // protect s_x/s_z before next tile overwrites them
    }
}

extern "C" void kernel_launch(void* const* d_in, const int* in_sizes, int n_in,
                              void* d_out, int out_size, void* d_ws, size_t ws_size,
                              hipStream_t stream) {
    (void)in_sizes; (void)n_in; (void)out_size; (void)d_ws; (void)ws_size;
    const float* X   = (const float*)d_in[0];
    const int*   jdg = (const int*)d_in[1];
    const float* W1  = (const float*)d_in[2];
    const float* W1a = (const float*)d_in[3];
    const float* W2  = (const float*)d_in[4];
    const float* W2a = (const float*)d_in[5];
    const float* V   = (const float*)d_in[6];
    const float* Va  = (const float*)d_in[7];
    dim3 grid(JN, 4);   // 64 judges x 4 tile slices = 256 workgroups, 8 waves each
    pcn_kernel<<<grid, 256, 0, stream>>>(X, jdg, W1, W1a, W2, W2a, V, Va,
                                         (float*)d_out);
}